// ParameterizedMultiHopCenterEmbedding_1735166788587
// MI455X (gfx1250) — compile-verified
//
#include <hip/hip_runtime.h>

#define HOPS   3
#define LABELS 10
#define LPAD   16      // labels padded to WMMA M=16
#define DIM    128
#define TILES  8       // 128 / 16

typedef __attribute__((ext_vector_type(2))) float v2f;
typedef __attribute__((ext_vector_type(8))) float v8f;

// Workspace layout (d_ws):
//   [0 .. 3*16*128)  float  per-hop padded label sums
//   [6144 .. 6160)   uint   per-label counts (padded to 16)

__global__ void zero_ws_k(float* ws, int total) {
  int i = blockIdx.x * blockDim.x + threadIdx.x;
  if (i < total) ws[i] = 0.0f;   // bit pattern 0 also zeroes the uint counts
}

__global__ void count_k(const int* __restrict__ labels, int n,
                        unsigned int* __restrict__ counts) {
  __shared__ unsigned int lds[LPAD];
  if (threadIdx.x < LPAD) lds[threadIdx.x] = 0u;
  __syncthreads();
  for (int i = blockIdx.x * blockDim.x + threadIdx.x; i < n;
       i += gridDim.x * blockDim.x) {
    atomicAdd(&lds[labels[i] & (LPAD - 1)], 1u);
  }
  __syncthreads();
  if (threadIdx.x < LPAD) {
    unsigned int c = lds[threadIdx.x];
    if (c) atomicAdd(&counts[threadIdx.x], c);
  }
}

// Segment-sum as one-hot GEMM on the fp32 matrix unit:
// sums[16, 128] += onehot(labels[g..g+3])^T(16x4) @ emb[g..g+3, :](4x128)
__global__ void segsum_wmma_k(const float* __restrict__ emb,
                              const int* __restrict__ labels,
                              float* __restrict__ sums, int n) {
  const int hop = blockIdx.y;
  const float* e = emb + (size_t)hop * (size_t)n * DIM;
  float* outp = sums + hop * (LPAD * DIM);

  __shared__ float lsum[LPAD * DIM];
  for (int i = threadIdx.x; i < LPAD * DIM; i += blockDim.x) lsum[i] = 0.0f;
  __syncthreads();

  const int lane = threadIdx.x & 31;
  const int wave = threadIdx.x >> 5;
  const int wavesPerBlock = blockDim.x >> 5;
  const int gw = blockIdx.x * wavesPerBlock + wave;   // global wave id (per hop)
  const int tw = gridDim.x * wavesPerBlock;           // total waves (per hop)
  const int half = lane >> 4;   // 0: lanes 0-15, 1: lanes 16-31
  const int col  = lane & 15;   // row index m for A, column n for B/C

  const int nFull = n & ~3;     // guard-free region (groups of 4 nodes)

  v8f acc[TILES];
#pragma unroll
  for (int t = 0; t < TILES; ++t) acc[t] = (v8f)0.0f;

  // Wave-uniform, guard-free hot loop (EXEC all-ones around the WMMAs).
  for (int g = gw * 4; g < nFull; g += tw * 4) {
    // ISA layout: for both A(16x4) and B(4x16) f32, VGPR j holds K = 2*half + j.
    const int n0 = g + 2 * half;                 // node for vgpr 0; vgpr 1 = n0+1
    const int l0 = labels[n0];
    const int l1 = labels[n0 + 1];
    v2f a;
    a.x = (l0 == col) ? 1.0f : 0.0f;
    a.y = (l1 == col) ? 1.0f : 0.0f;
    // Single base: node n0 row at +0, node n0+1 row at +DIM floats (+512 B).
    const float* r = e + (size_t)n0 * DIM + col;
#pragma unroll
    for (int t = 0; t < TILES; ++t) {
      v2f b;
      b.x = r[t * 16];
      b.y = r[DIM + t * 16];
      acc[t] = __builtin_amdgcn_wmma_f32_16x16x4_f32(
          /*neg_a=*/false, a, /*neg_b=*/false, b,
          /*c_mod=*/(short)0, acc[t], /*reuse_a=*/false, /*reuse_b=*/false);
    }
  }

  // C/D layout: VGPR v, lanes(half) -> row m = v + 8*half, column = col.
#pragma unroll
  for (int t = 0; t < TILES; ++t) {
#pragma unroll
    for (int v = 0; v < 8; ++v) {
      atomicAdd(&lsum[(v + 8 * half) * DIM + t * 16 + col], acc[t][v]);
    }
  }

  // Tail nodes (n % 4 != 0): scalar path, one wave per hop handles <= 3 nodes.
  if (blockIdx.x == 0 && wave == 0) {
    for (int tn = nFull; tn < n; ++tn) {
      const int l = labels[tn];
      const float* r = e + (size_t)tn * DIM;
      for (int d = lane; d < DIM; d += 32) {
        atomicAdd(&lsum[l * DIM + d], r[d]);
      }
    }
  }

  __syncthreads();
  for (int i = threadIdx.x; i < LPAD * DIM; i += blockDim.x) {
    float s = lsum[i];
    if (s != 0.0f) atomicAdd(&outp[i], s);
  }
}

__global__ void finalize_k(const float* __restrict__ sums,
                           const unsigned int* __restrict__ counts,
                           const float* __restrict__ weight,
                           float* __restrict__ out) {
  int i = blockIdx.x * blockDim.x + threadIdx.x;
  const int NC = HOPS * LABELS * DIM;
  if (i < NC) {
    int d = i & (DIM - 1);
    int l = (i / DIM) % LABELS;
    int h = i / (LABELS * DIM);
    float c = (float)counts[l];
    float safe = c > 1.0f ? c : 1.0f;
    out[i] = sums[h * LPAD * DIM + l * DIM + d] / safe + weight[i];
  } else if (i < NC + LABELS) {
    out[i] = (float)counts[i - NC];   // class_counts, flattened [10,1]
  }
}

extern "C" void kernel_launch(void* const* d_in, const int* in_sizes, int n_in,
                              void* d_out, int out_size, void* d_ws, size_t ws_size,
                              hipStream_t stream) {
  const float* emb    = (const float*)d_in[0];  // [3, N, 128] f32
  const int*   labels = (const int*)d_in[1];    // [N] int
  const float* weight = (const float*)d_in[2];  // [3, 10, 128] f32
  float* out = (float*)d_out;                   // 3*10*128 centers+w, then 10 counts
  const int n = in_sizes[1];

  float* sums = (float*)d_ws;
  unsigned int* counts = (unsigned int*)(sums + HOPS * LPAD * DIM);

  const int zn = HOPS * LPAD * DIM + LPAD;
  zero_ws_k<<<(zn + 255) / 256, 256, 0, stream>>>((float*)d_ws, zn);
  count_k<<<256, 256, 0, stream>>>(labels, n, counts);
  dim3 grid(128, HOPS);
  segsum_wmma_k<<<grid, 256, 0, stream>>>(emb, labels, sums, n);
  const int fn = HOPS * LABELS * DIM + LABELS;
  finalize_k<<<(fn + 255) / 256, 256, 0, stream>>>(sums, counts, weight, out);
}